// DeepGMM_12575664243257
// MI455X (gfx1250) — compile-verified
//
#include <hip/hip_runtime.h>
#include <math.h>

// ---------------- problem constants ----------------
#define NN   384
#define SS   10
#define KK   32
#define XD   64
#define YD   64
#define REGC 0.01f
#define LOG2PI 1.8378770664093453f

// ---------------- workspace layout (float offsets) ----------------
static constexpr size_t O_ENCMU = 0;                       // [N,64]
static constexpr size_t O_ENCD  = O_ENCMU + (size_t)NN*64; // [N,64]
static constexpr size_t O_ZLP   = O_ENCD  + (size_t)NN*64; // [N,K] logits -> log_probs
static constexpr size_t O_DT    = O_ZLP   + (size_t)NN*KK; // [N,K,64] Sigma_tilde diag
static constexpr size_t O_W     = O_DT    + (size_t)NN*KK*64; // [N,K,64] rank-1 vec
static constexpr size_t O_GAM   = O_W     + (size_t)NN*KK*64; // [N,K]
static constexpr size_t O_MUT   = O_GAM   + (size_t)NN*KK;    // [N,K,64]
static constexpr size_t O_ZS    = O_MUT   + (size_t)NN*KK*64; // [N,S,K]
static constexpr size_t O_XS    = O_ZS    + (size_t)NN*SS*KK; // [N*S,64]
static constexpr size_t O_MUY   = O_XS    + (size_t)NN*SS*64; // [N*S,64]
static constexpr size_t O_DD    = O_MUY   + (size_t)NN*SS*64; // [N*S,64]
static constexpr size_t O_ACC   = O_DD    + (size_t)NN*SS*64; // [8] accumulators

// ---------------- small helpers ----------------
typedef __attribute__((ext_vector_type(2))) float v2f;
typedef __attribute__((ext_vector_type(8))) float v8f;

__device__ inline float wred(float v) {
    for (int o = 16; o > 0; o >>= 1) v += __shfl_xor(v, o, 32);
    return v;
}
__device__ inline float wmaxr(float v) {
    for (int o = 16; o > 0; o >>= 1) v = fmaxf(v, __shfl_xor(v, o, 32));
    return v;
}
__device__ inline float sigm(float x) { return 1.0f / (1.0f + expf(-x)); }

__device__ inline float block_sum(float v, volatile float* red, int tid) {
    int lane = tid & 31, wv = tid >> 5;
    for (int o = 16; o > 0; o >>= 1) v += __shfl_xor(v, o, 32);
    if (lane == 0) red[wv] = v;
    __syncthreads();
    float s = red[0] + red[1] + red[2] + red[3] + red[4] + red[5] + red[6] + red[7];
    __syncthreads();
    return s;
}

// Build Sig(64x64) = Fs(64x32) @ Fm^T(32x64) using V_WMMA_F32_16X16X4_F32.
// A layout (16x4 f32): lanes 0-15 rows M=0..15 hold K={k0,k0+1}; lanes 16-31 hold K={k0+2,k0+3}.
// B layout (4x16 f32): VGPR0 = row k0 (lanes 0-15) / row k0+2 (lanes 16-31); VGPR1 = k0+1 / k0+3.
// C/D layout (16x16 f32, 8 VGPRs): VGPR r -> M=r (lanes 0-15) / M=r+8 (lanes 16-31), N = lane&15.
__device__ inline void wmma_build64(const float* Fs, const float* Fm, float* Sig,
                                    int wv, int lane) {
    const int half = lane >> 4;
    const int l15  = lane & 15;
    for (int t = wv; t < 16; t += 8) {
        const int ti = t >> 2, tj = t & 3;
        v8f c = {0.f, 0.f, 0.f, 0.f, 0.f, 0.f, 0.f, 0.f};
        const int arow = ti * 16 + l15;
        const int bcol = tj * 16 + l15;
        for (int k0 = 0; k0 < 32; k0 += 4) {
            const int ka = k0 + half * 2;
            v2f a, b;
            a.x = Fs[arow * 32 + ka];
            a.y = Fs[arow * 32 + ka + 1];
            b.x = Fm[bcol * 32 + ka];      // B[ka][bcol] = Fm^T
            b.y = Fm[bcol * 32 + ka + 1];
            c = __builtin_amdgcn_wmma_f32_16x16x4_f32(false, a, false, b,
                                                      (short)0, c, false, false);
        }
        for (int r = 0; r < 8; r++) {
            const int row = ti * 16 + r + half * 8;
            Sig[row * 64 + tj * 16 + l15] = c[r];
        }
    }
}

// cooperative in-place lower Cholesky of 64x64 in LDS; ends synchronized
__device__ inline void chol64(float* A, int tid) {
    for (int j = 0; j < 64; j++) {
        if (tid == 0) {
            float s = A[j * 64 + j];
            for (int p = 0; p < j; p++) { float v = A[j * 64 + p]; s -= v * v; }
            A[j * 64 + j] = sqrtf(fmaxf(s, 1e-12f));
        }
        __syncthreads();
        const int i = j + 1 + tid;
        if (i < 64) {
            float s = A[i * 64 + j];
            for (int p = 0; p < j; p++) s -= A[i * 64 + p] * A[j * 64 + p];
            A[i * 64 + j] = s / A[j * 64 + j];
        }
        __syncthreads();
    }
}

// forward solve L y = y in place; ends synchronized
__device__ inline void fsolve64(const float* L, float* y, int tid) {
    for (int j = 0; j < 64; j++) {
        if (tid == 0) y[j] = y[j] / L[j * 64 + j];
        __syncthreads();
        const int i = j + 1 + tid;
        if (i < 64) y[i] -= L[i * 64 + j] * y[j];
        __syncthreads();
    }
}

// ---------------- kernel 0: zero accumulators ----------------
__global__ void zero_acc_kernel(float* ws) {
    if (threadIdx.x < 8) ws[O_ACC + threadIdx.x] = 0.0f;
}

// ---------------- kernel 1/6: C = A[M,64] @ {Wmu|Ws}[64,64] + bias, second half sigmoid+reg
// grid (M/16, 8), block 32 (one wave per 16x16 output tile), WMMA f32 16x16x4
__global__ __launch_bounds__(32) void gemm_bias_kernel(
    const float* __restrict__ A,
    const float* __restrict__ Wmu, const float* __restrict__ bmu,
    const float* __restrict__ Wsg, const float* __restrict__ bsg,
    float* __restrict__ Omu, float* __restrict__ Osg, float reg)
{
    const int ti = blockIdx.x, tj = blockIdx.y;
    const int lane = threadIdx.x;
    const bool second = tj >= 4;
    const float* W    = second ? Wsg : Wmu;
    const float* bias = second ? bsg : bmu;
    float* O          = second ? Osg : Omu;
    const int cbase = (second ? (tj - 4) : tj) * 16;
    const int half = lane >> 4, l15 = lane & 15;
    const int arow = ti * 16 + l15;
    v8f c = {0.f, 0.f, 0.f, 0.f, 0.f, 0.f, 0.f, 0.f};
    for (int k0 = 0; k0 < 64; k0 += 4) {
        const int ka = k0 + half * 2;
        v2f a, b;
        a.x = A[arow * 64 + ka];
        a.y = A[arow * 64 + ka + 1];
        const int col = cbase + l15;
        b.x = W[ka * 64 + col];
        b.y = W[(ka + 1) * 64 + col];
        c = __builtin_amdgcn_wmma_f32_16x16x4_f32(false, a, false, b,
                                                  (short)0, c, false, false);
    }
    for (int r = 0; r < 8; r++) {
        const int row = ti * 16 + r + half * 8;
        const int col = cbase + l15;
        float v = c[r] + bias[col];
        if (second) v = sigm(v) + reg;
        O[row * 64 + col] = v;
    }
}

// ---------------- kernel 2: per-(n,k) analytic posterior (one wave per pair) ---------
__global__ __launch_bounds__(32) void cluster_post_kernel(
    float* __restrict__ ws,
    const float* __restrict__ phi_mus, const float* __restrict__ phi_sd,
    const float* __restrict__ phi_f,  const float* __restrict__ phi_logits)
{
    const int nk = blockIdx.x;
    const int n = nk >> 5, k = nk & 31;
    const int l = threadIdx.x;
    // log_softmax(phi_logits)[k]  (K == 32 == wave width)
    {
        float lg = phi_logits[l];
        float m  = wmaxr(lg);
        float se = wred(expf(lg - m));
        // fall through; all lanes hold same m, se
        float logpi = phi_logits[k] - (m + logf(se));

        const int d0 = l, d1 = l + 32;
        const float e0 = ws[O_ENCD + (size_t)n * 64 + d0];
        const float e1 = ws[O_ENCD + (size_t)n * 64 + d1];
        const float em0 = ws[O_ENCMU + (size_t)n * 64 + d0];
        const float em1 = ws[O_ENCMU + (size_t)n * 64 + d1];
        const float pm0 = phi_mus[k * 64 + d0], pm1 = phi_mus[k * 64 + d1];
        const float p0 = sigm(phi_sd[k * 64 + d0]) + REGC;
        const float p1 = sigm(phi_sd[k * 64 + d1]) + REGC;
        const float f0 = phi_f[k * 64 + d0], f1 = phi_f[k * 64 + d1];
        const float a0 = e0 + p0, a1 = e1 + p1;
        const float df0 = em0 - pm0, df1 = em1 - pm1;

        // z-logit via Sherman-Morrison + determinant lemma on diag(a)+f f^T
        const float s1 = wred(f0 * f0 / a0 + f1 * f1 / a1);
        const float s2 = wred(df0 * df0 / a0 + df1 * df1 / a1);
        const float s3 = wred(df0 * f0 / a0 + df1 * f1 / a1);
        const float sld = wred(logf(a0) + logf(a1));
        const float Mq = s2 - s3 * s3 / (1.0f + s1);
        const float zlogit = logpi - 0.5f * (64.0f * LOG2PI + Mq)
                                   - 0.5f * (sld + log1pf(s1));

        // Sigma_tilde = (diag(1/e) + inv(diag(p)+f f^T))^{-1} = diag(1/q) + gam*w w^T
        const float u0 = f0 / p0, u1 = f1 / p1;
        const float s4 = wred(f0 * u0 + f1 * u1);
        const float beta = 1.0f / (1.0f + s4);
        const float q0 = 1.0f / e0 + 1.0f / p0, q1 = 1.0f / e1 + 1.0f / p1;
        const float w0 = u0 / q0, w1 = u1 / q1;
        const float s5 = wred(u0 * w0 + u1 * w1);
        const float gam = beta / (1.0f - beta * s5);
        const float s6 = wred(u0 * pm0 + u1 * pm1);
        const float r0 = em0 / e0 + pm0 / p0 - beta * u0 * s6;
        const float r1 = em1 / e1 + pm1 / p1 - beta * u1 * s6;
        const float s7 = wred(w0 * r0 + w1 * r1);
        const float mt0 = r0 / q0 + gam * w0 * s7;
        const float mt1 = r1 / q1 + gam * w1 * s7;

        const size_t base = (size_t)nk * 64;
        ws[O_DT + base + d0] = 1.0f / q0;  ws[O_DT + base + d1] = 1.0f / q1;
        ws[O_W  + base + d0] = w0;         ws[O_W  + base + d1] = w1;
        ws[O_MUT + base + d0] = mt0;       ws[O_MUT + base + d1] = mt1;
        if (l == 0) { ws[O_GAM + nk] = gam; ws[O_ZLP + nk] = zlogit; }
    }
}

// ---------------- kernel 3: log_softmax of z_logits per n (+ loss5) ----------------
__global__ __launch_bounds__(32) void zsoftmax_kernel(float* __restrict__ ws)
{
    const int n = blockIdx.x, l = threadIdx.x;
    float v  = ws[O_ZLP + (size_t)n * 32 + l];
    float m  = wmaxr(v);
    float se = wred(expf(v - m));
    float lp = v - (m + logf(se));
    ws[O_ZLP + (size_t)n * 32 + l] = lp;
    float se2 = wred(expf(lp));
    if (l == 0) atomicAdd(&ws[O_ACC + 2], logf(se2));
}

// ---------------- kernel 4: Gumbel-softmax samples per (n,s) ----------------
__global__ __launch_bounds__(32) void gumbel_kernel(float* __restrict__ ws,
                                                    const float* __restrict__ u_noise)
{
    const int ns = blockIdx.x, l = threadIdx.x;
    const int n = ns / SS;
    const float g = -logf(-logf(u_noise[(size_t)ns * 32 + l]));
    const float v = ws[O_ZLP + (size_t)n * 32 + l] + g;   // TEMP == 1
    const float m = wmaxr(v);
    const float e = expf(v - m);
    const float se = wred(e);
    ws[O_ZS + (size_t)ns * 32 + l] = e / se;
}

// ---------------- kernel 5: fused per-sample mix + Cholesky + losses 2/3/4 ----------
__global__ __launch_bounds__(256) void sample_loss_kernel(
    float* __restrict__ ws,
    const float* __restrict__ phi_mus, const float* __restrict__ phi_sd,
    const float* __restrict__ phi_f,
    const float* __restrict__ th_mus,  const float* __restrict__ th_sd,
    const float* __restrict__ th_f,    const float* __restrict__ th_logits,
    const float* __restrict__ eps)
{
    __shared__ float Sig[64 * 64];
    __shared__ float Fm[64 * 32];
    __shared__ float Fs[64 * 32];
    __shared__ float zv[32], zg[32], dia[64], muv[64], xv[64], yv[64];
    __shared__ float tls[32], red[8], scal[2];

    const int ns = blockIdx.x;
    const int n = ns / SS;
    const int tid = threadIdx.x, lane = tid & 31, wv = tid >> 5;

    if (tid < 32) zv[tid] = ws[O_ZS + (size_t)ns * 32 + tid];
    if (wv == 0) {  // log_softmax(theta_logits) (wave-local; zv written by this wave)
        float v = th_logits[lane];
        float m = wmaxr(v);
        float se = wred(expf(v - m));
        tls[lane] = v - (m + logf(se));
    }
    __syncthreads();

    if (wv == 0) {  // dot(log_softmax(theta_logits), z) and dot(z_log_probs[n], z)
        float a = tls[lane] * zv[lane];
        float b = ws[O_ZLP + (size_t)n * 32 + lane] * zv[lane];
        a = wred(a); b = wred(b);
        if (lane == 0) { scal[0] = a; scal[1] = b; }
    }

    // ---- mix Sigma_tilde pieces: dia = sum_k z dt, muv = sum_k z mu_tilde ----
    if (tid < 64) {
        const float* dtp = ws + O_DT  + (size_t)n * (KK * 64);
        const float* mtp = ws + O_MUT + (size_t)n * (KK * 64);
        float dsum = 0.f, msum = 0.f;
        for (int k = 0; k < 32; k++) {
            const float z = zv[k];
            dsum += z * dtp[k * 64 + tid];
            msum += z * mtp[k * 64 + tid];
        }
        dia[tid] = dsum; muv[tid] = msum;
    }
    if (tid < 32) zg[tid] = zv[tid] * ws[O_GAM + (size_t)n * 32 + tid];
    __syncthreads();

    {   // Fm[i][k] = w[n,k,i] ; Fs = column-scaled by z*gamma
        const float* wvp = ws + O_W + (size_t)n * (KK * 64);
        for (int e2 = tid; e2 < 2048; e2 += 256) {
            const int i = e2 >> 5, k = e2 & 31;
            const float val = wvp[k * 64 + i];
            Fm[e2] = val; Fs[e2] = zg[k] * val;
        }
    }
    __syncthreads();
    wmma_build64(Fs, Fm, Sig, wv, lane);   // Sig = Fm diag(z*gam) Fm^T
    __syncthreads();
    if (tid < 64) Sig[tid * 65] += dia[tid];
    __syncthreads();
    chol64(Sig, tid);

    // ---- x = mu_t + L @ eps ----
    if (tid < 64) yv[tid] = eps[(size_t)ns * 64 + tid];
    __syncthreads();
    if (tid < 64) {
        float acc = muv[tid];
        for (int j = 0; j <= tid; j++) acc += Sig[tid * 64 + j] * yv[j];
        xv[tid] = acc;
        ws[O_XS + (size_t)ns * 64 + tid] = acc;
    }
    __syncthreads();

    // ---- loss2 = -logN(x ; enc_mu, diag(enc_diag)) ----
    float t2 = 0.f;
    if (tid < 64) {
        const float e = ws[O_ENCD + (size_t)n * 64 + tid];
        const float d = xv[tid] - ws[O_ENCMU + (size_t)n * 64 + tid];
        t2 = d * d / e + logf(e);
    }
    const float s2s = block_sum(t2, red, tid);

    // ---- loss3: logN(x ; th_mu, th_Sig) with th_Sig = diag + F z F^T ----
    if (tid < 64) {
        float dsum = 0.f, msum = 0.f;
        for (int k = 0; k < 32; k++) {
            const float z = zv[k];
            dsum += z * (sigm(th_sd[k * 64 + tid]) + 2.0f * REGC);
            msum += z * th_mus[k * 64 + tid];
        }
        dia[tid] = dsum; muv[tid] = msum;
    }
    for (int e2 = tid; e2 < 2048; e2 += 256) {
        const int i = e2 >> 5, k = e2 & 31;
        const float val = th_f[k * 64 + i];
        Fm[e2] = val; Fs[e2] = zv[k] * val;
    }
    __syncthreads();
    wmma_build64(Fs, Fm, Sig, wv, lane);
    __syncthreads();
    if (tid < 64) Sig[tid * 65] += dia[tid];
    __syncthreads();
    chol64(Sig, tid);
    if (tid < 64) yv[tid] = xv[tid] - muv[tid];
    __syncthreads();
    fsolve64(Sig, yv, tid);
    float t3 = 0.f;
    if (tid < 64) t3 = yv[tid] * yv[tid] + 2.0f * logf(Sig[tid * 65]);
    const float s3s = block_sum(t3, red, tid);

    // ---- loss4: logN(x ; ph_mu, ph_Sig) with ph_Sig = diag + F z F^T ----
    if (tid < 64) {
        float dsum = 0.f, msum = 0.f;
        for (int k = 0; k < 32; k++) {
            const float z = zv[k];
            dsum += z * (sigm(phi_sd[k * 64 + tid]) + 2.0f * REGC);  // phi_diag + REG
            msum += z * phi_mus[k * 64 + tid];
        }
        dia[tid] = dsum; muv[tid] = msum;
    }
    for (int e2 = tid; e2 < 2048; e2 += 256) {
        const int i = e2 >> 5, k = e2 & 31;
        const float val = phi_f[k * 64 + i];
        Fm[e2] = val; Fs[e2] = zv[k] * val;
    }
    __syncthreads();
    wmma_build64(Fs, Fm, Sig, wv, lane);
    __syncthreads();
    if (tid < 64) Sig[tid * 65] += dia[tid];
    __syncthreads();
    chol64(Sig, tid);
    if (tid < 64) yv[tid] = xv[tid] - muv[tid];
    __syncthreads();
    fsolve64(Sig, yv, tid);
    float t4 = 0.f;
    if (tid < 64) t4 = yv[tid] * yv[tid] + 2.0f * logf(Sig[tid * 65]);
    const float s4s = block_sum(t4, red, tid);

    if (tid == 0) {
        const float loss2 = 0.5f * (64.0f * LOG2PI + s2s);
        const float loss3 = -0.5f * (64.0f * LOG2PI + s3s) + scal[0];
        const float loss4 = -(-0.5f * (64.0f * LOG2PI + s4s) + scal[1]);
        atomicAdd(&ws[O_ACC + 0], loss2 + loss3 + loss4);
    }
}

// ---------------- kernel 7: loss1 (diagonal decoder likelihood) ----------------
__global__ __launch_bounds__(32) void loss1_kernel(float* __restrict__ ws,
                                                   const float* __restrict__ Y)
{
    const int ns = blockIdx.x, l = threadIdx.x;
    const int n = ns / SS;
    float t = 0.f;
    for (int h = 0; h < 2; h++) {
        const int d = l + h * 32;
        const float dd = ws[O_DD + (size_t)ns * 64 + d];
        const float df = Y[(size_t)n * 64 + d] - ws[O_MUY + (size_t)ns * 64 + d];
        t += df * df / dd + logf(dd);
    }
    const float s = wred(t);
    if (l == 0) atomicAdd(&ws[O_ACC + 1], -0.5f * (64.0f * LOG2PI + s));
}

// ---------------- kernel 8: final combine ----------------
__global__ void final_kernel(const float* __restrict__ ws, float* __restrict__ out)
{
    if (threadIdx.x == 0)
        out[0] = -((ws[O_ACC + 0] + ws[O_ACC + 1]) / (float)SS + ws[O_ACC + 2]);
}

// ---------------- host launcher ----------------
extern "C" void kernel_launch(void* const* d_in, const int* in_sizes, int n_in,
                              void* d_out, int out_size, void* d_ws, size_t ws_size,
                              hipStream_t stream)
{
    const float* Y        = (const float*)d_in[0];
    const float* W_enc_mu = (const float*)d_in[1];
    const float* b_enc_mu = (const float*)d_in[2];
    const float* W_enc_s  = (const float*)d_in[3];
    const float* b_enc_s  = (const float*)d_in[4];
    const float* W_dec_mu = (const float*)d_in[5];
    const float* b_dec_mu = (const float*)d_in[6];
    const float* W_dec_s  = (const float*)d_in[7];
    const float* b_dec_s  = (const float*)d_in[8];
    const float* phi_mus  = (const float*)d_in[9];
    const float* phi_sd   = (const float*)d_in[10];
    const float* phi_f    = (const float*)d_in[11];
    const float* phi_log  = (const float*)d_in[12];
    const float* th_mus   = (const float*)d_in[13];
    const float* th_sd    = (const float*)d_in[14];
    const float* th_f     = (const float*)d_in[15];
    const float* th_log   = (const float*)d_in[16];
    const float* u_noise  = (const float*)d_in[17];
    const float* eps      = (const float*)d_in[18];
    float* ws  = (float*)d_ws;
    float* out = (float*)d_out;

    zero_acc_kernel<<<1, 32, 0, stream>>>(ws);
    // encoder: Y[384,64] @ W_enc_{mu,s} (+bias, sigmoid+reg on s)
    gemm_bias_kernel<<<dim3(NN / 16, 8), 32, 0, stream>>>(
        Y, W_enc_mu, b_enc_mu, W_enc_s, b_enc_s,
        ws + O_ENCMU, ws + O_ENCD, REGC);
    // per-(n,k) analytic Gaussian fusion
    cluster_post_kernel<<<NN * KK, 32, 0, stream>>>(ws, phi_mus, phi_sd, phi_f, phi_log);
    // z log-softmax + loss5
    zsoftmax_kernel<<<NN, 32, 0, stream>>>(ws);
    // gumbel-softmax samples
    gumbel_kernel<<<NN * SS, 32, 0, stream>>>(ws, u_noise);
    // fused per-sample mixing + Cholesky + losses 2/3/4
    sample_loss_kernel<<<NN * SS, 256, 0, stream>>>(
        ws, phi_mus, phi_sd, phi_f, th_mus, th_sd, th_f, th_log, eps);
    // decoder: x[3840,64] @ W_dec_{mu,s}
    gemm_bias_kernel<<<dim3(NN * SS / 16, 8), 32, 0, stream>>>(
        ws + O_XS, W_dec_mu, b_dec_mu, W_dec_s, b_dec_s,
        ws + O_MUY, ws + O_DD, REGC);
    // loss1
    loss1_kernel<<<NN * SS, 32, 0, stream>>>(ws, Y);
    // final scalar
    final_kernel<<<1, 32, 0, stream>>>(ws, out);
}